// VGAEEncoder_4483945857665
// MI455X (gfx1250) — compile-verified
//
#include <hip/hip_runtime.h>

// ---------------------------------------------------------------------------
// VGAE encoder (2x GCNConv + GCN heads) for MI455X / gfx1250 (wave32, WMMA).
// Dense transforms use V_WMMA_F32_16X16X4_F32; weight staging uses the CDNA5
// async global->LDS path when available; sparse aggregation uses global f32
// atomics (L2-resident working set: everything fits in the 192MB L2).
// ---------------------------------------------------------------------------

typedef __attribute__((ext_vector_type(2))) float v2f;
typedef __attribute__((ext_vector_type(8))) float v8f;

#define NNODES_INDIM 256

#if defined(__gfx1250__) && \
    __has_builtin(__builtin_amdgcn_global_load_async_to_lds_b32) && \
    __has_builtin(__builtin_amdgcn_s_wait_asynccnt)
#define HAVE_ASYNC_LDS 1
typedef __attribute__((address_space(1))) int gas_int;  // printed as __device__ int*
typedef __attribute__((address_space(3))) int lds_int;  // printed as __shared__ int*
#else
#define HAVE_ASYNC_LDS 0
#endif

// ---------------- degree / norm kernels ----------------

__global__ void deg_init_k(float* __restrict__ deg, int n) {
  int i = blockIdx.x * blockDim.x + threadIdx.x;
  if (i < n) deg[i] = 1.0f;  // self-loop contribution
}

__global__ void deg_edge_k(const int* __restrict__ dst, float* __restrict__ deg, int nE) {
  int e = blockIdx.x * blockDim.x + threadIdx.x;
  if (e < nE) atomicAdd(&deg[dst[e]], 1.0f);
}

__global__ void deg_rsqrt_k(float* __restrict__ deg, int n) {
  int i = blockIdx.x * blockDim.x + threadIdx.x;
  if (i < n) deg[i] = rsqrtf(fmaxf(deg[i], 1.0f));  // deg -> dinv in place
}

// ---------------- WMMA f32 GEMM: out[n,NDIM] = A[n,KDIM] @ W[KDIM,NDIM] -----
// One wave computes a 16-row M tile; 8 waves (256 threads) per block.
// Requires n % 16 == 0 (true here: 100000 = 6250 * 16).

template <int KDIM, int NDIM>
__global__ __launch_bounds__(256) void wmma_gemm_f32_k(
    const float* __restrict__ A, const float* __restrict__ W,
    float* __restrict__ out, int n) {
  constexpr int NT = NDIM / 16;  // number of 16-wide N tiles
  __shared__ float w[KDIM * NDIM];

#if HAVE_ASYNC_LDS
  // CDNA5 async global->LDS copy of the weight tile (ASYNCcnt-tracked).
  for (int i = threadIdx.x; i < KDIM * NDIM; i += 256) {
    __builtin_amdgcn_global_load_async_to_lds_b32(
        (gas_int*)(size_t)(W + i),
        (lds_int*)(unsigned)(size_t)(&w[i]),
        /*offset=*/0, /*cpol=*/0);
  }
  __builtin_amdgcn_s_wait_asynccnt(0);
  __syncthreads();
#else
  for (int i = threadIdx.x; i < KDIM * NDIM; i += 256) w[i] = W[i];
  __syncthreads();
#endif

  const int lane   = threadIdx.x & 31;
  const int wave   = threadIdx.x >> 5;
  const int laneLo = lane & 15;
  const int hi     = lane >> 4;  // 0: lanes 0-15, 1: lanes 16-31

  const long m0 = ((long)blockIdx.x * 8 + wave) * 16;
  if (m0 >= n) return;

  v8f acc[NT];
#pragma unroll
  for (int t = 0; t < NT; ++t) {
    v8f z = {0.f, 0.f, 0.f, 0.f, 0.f, 0.f, 0.f, 0.f};
    acc[t] = z;
  }

  // A fragment: lane holds row M = m0 + laneLo, K = k + 2*hi + {0,1}
  const float* arow = A + (m0 + laneLo) * KDIM + 2 * hi;

  for (int k0 = 0; k0 < KDIM; k0 += 32) {
    // pull the next 128B of this lane's row ahead of the strided loads
    __builtin_prefetch(arow + k0 + 32, 0, 0);
#pragma unroll
    for (int kk = 0; kk < 32; kk += 4) {
      const int k = k0 + kk;
      v2f a = *(const v2f*)(arow + k);
      // B fragment: VGPR0 = row K=k (lanes 0-15) / K=k+2 (lanes 16-31),
      //             VGPR1 = row K=k+1 / K=k+3; column N = ntile*16 + laneLo.
      const float* wk = &w[(k + 2 * hi) * NDIM + laneLo];
#pragma unroll
      for (int t = 0; t < NT; ++t) {
        v2f b;
        b.x = wk[t * 16];
        b.y = wk[NDIM + t * 16];
        acc[t] = __builtin_amdgcn_wmma_f32_16x16x4_f32(
            /*neg_a=*/false, a, /*neg_b=*/false, b,
            /*c_mod=*/(short)0, acc[t], /*reuse_a=*/false, /*reuse_b=*/false);
      }
    }
  }

  // D layout: VGPR v -> row M = m0 + v + 8*hi, col = ntile*16 + laneLo
  const long row = m0 + 8 * hi;
#pragma unroll
  for (int v = 0; v < 8; ++v) {
    float* orow = out + (row + v) * NDIM + laneLo;
#pragma unroll
    for (int t = 0; t < NT; ++t) orow[t * 16] = acc[t][v];
  }
}

// ---------------- GEMV head: out[n] = h[n,32] @ w3[32] ----------------

__global__ void gemv32_k(const float* __restrict__ h, const float* __restrict__ w3,
                         float* __restrict__ out, int n) {
  int i = blockIdx.x * blockDim.x + threadIdx.x;
  if (i >= n) return;
  const float* r = h + (long)i * 32;
  float s = 0.f;
#pragma unroll
  for (int k = 0; k < 32; ++k) s = fmaf(r[k], w3[k], s);
  out[i] = s;
}

// ---------------- aggregation: out = D^-1/2 (A+I) D^-1/2 t + b -------------

// self-loop term + bias (also initializes the accumulator buffer; no zero fill needed)
template <int F>
__global__ void agg_init_k(const float* __restrict__ t, const float* __restrict__ dinv,
                           const float* __restrict__ bias, float* __restrict__ out, int n) {
  long g = (long)blockIdx.x * blockDim.x + threadIdx.x;
  if (g >= (long)n * F) return;
  int i = (int)(g / F);
  int f = (int)(g % F);
  float dv = dinv[i];
  out[g] = t[g] * dv * dv + bias[f];
}

template <int F>
__global__ void agg_edge_k(const float* __restrict__ t, const float* __restrict__ dinv,
                           const int* __restrict__ src, const int* __restrict__ dst,
                           float* __restrict__ out, int nE) {
  long g = (long)blockIdx.x * blockDim.x + threadIdx.x;
  if (g >= (long)nE * F) return;
  int e = (int)(g / F);
  int f = (int)(g % F);
  int s = src[e];
  int d = dst[e];
  float nrm = dinv[s] * dinv[d];
  atomicAdd(&out[(long)d * F + f], t[(long)s * F + f] * nrm);
}

__global__ void agg_init1_k(const float* __restrict__ t, const float* __restrict__ dinv,
                            const float* __restrict__ bias, float* __restrict__ out, int n) {
  int i = blockIdx.x * blockDim.x + threadIdx.x;
  if (i >= n) return;
  float dv = dinv[i];
  out[i] = t[i] * dv * dv + bias[0];
}

__global__ void agg_edge1_k(const float* __restrict__ t, const float* __restrict__ dinv,
                            const int* __restrict__ src, const int* __restrict__ dst,
                            float* __restrict__ out, int nE) {
  int e = blockIdx.x * blockDim.x + threadIdx.x;
  if (e >= nE) return;
  int s = src[e];
  int d = dst[e];
  atomicAdd(&out[d], t[s] * dinv[s] * dinv[d]);
}

__global__ void relu_k(float* __restrict__ x, long total) {
  long g = (long)blockIdx.x * blockDim.x + threadIdx.x;
  if (g < total) x[g] = fmaxf(x[g], 0.f);
}

// ---------------------------------------------------------------------------

extern "C" void kernel_launch(void* const* d_in, const int* in_sizes, int n_in,
                              void* d_out, int out_size, void* d_ws, size_t ws_size,
                              hipStream_t stream) {
  (void)n_in; (void)out_size; (void)ws_size;
  const float* X  = (const float*)d_in[0];
  const int* src  = (const int*)d_in[1];
  const int* dst  = (const int*)d_in[2];
  const float* W1 = (const float*)d_in[3];
  const float* b1 = (const float*)d_in[4];
  const float* W2 = (const float*)d_in[5];
  const float* b2 = (const float*)d_in[6];
  const float* W3 = (const float*)d_in[7];
  const float* b3 = (const float*)d_in[8];
  float* out = (float*)d_out;

  const int n  = in_sizes[0] / NNODES_INDIM;  // 100000
  const int nE = in_sizes[1];                 // 1600000

  // workspace layout (floats): dinv[n] | t1[n*32] | h[n*32] | t2[n*64] | t3[n]
  float* dinv = (float*)d_ws;
  float* t1 = dinv + n;
  float* h  = t1 + (size_t)n * 32;
  float* t2 = h  + (size_t)n * 32;
  float* t3 = t2 + (size_t)n * 64;

  float* mus = out;                   // [n, 64]
  float* ls  = out + (size_t)n * 64;  // [n, 1]

  const int B = 256;
  const unsigned nb_n  = (unsigned)((n + B - 1) / B);
  const unsigned nb_e  = (unsigned)((nE + B - 1) / B);
  const unsigned nb_mm = (unsigned)((n / 16 + 7) / 8);  // 8 M-tiles per block

  // --- degree / normalization ---
  deg_init_k<<<nb_n, B, 0, stream>>>(dinv, n);
  deg_edge_k<<<nb_e, B, 0, stream>>>(dst, dinv, nE);
  deg_rsqrt_k<<<nb_n, B, 0, stream>>>(dinv, n);

  // --- layer 1: h = relu(agg(X @ W1) + b1) ---
  wmma_gemm_f32_k<256, 32><<<nb_mm, B, 0, stream>>>(X, W1, t1, n);
  agg_init_k<32><<<(unsigned)(((long)n * 32 + B - 1) / B), B, 0, stream>>>(t1, dinv, b1, h, n);
  agg_edge_k<32><<<(unsigned)(((long)nE * 32 + B - 1) / B), B, 0, stream>>>(t1, dinv, src, dst, h, nE);
  relu_k<<<(unsigned)(((long)n * 32 + B - 1) / B), B, 0, stream>>>(h, (long)n * 32);

  // --- layer 2: mus = agg(h @ W2) + b2 ---
  wmma_gemm_f32_k<32, 64><<<nb_mm, B, 0, stream>>>(h, W2, t2, n);
  agg_init_k<64><<<(unsigned)(((long)n * 64 + B - 1) / B), B, 0, stream>>>(t2, dinv, b2, mus, n);
  agg_edge_k<64><<<(unsigned)(((long)nE * 64 + B - 1) / B), B, 0, stream>>>(t2, dinv, src, dst, mus, nE);

  // --- layer 3: logsigma2s = agg(h @ W3) + b3 ---
  gemv32_k<<<nb_n, B, 0, stream>>>(h, W3, t3, n);
  agg_init1_k<<<nb_n, B, 0, stream>>>(t3, dinv, b3, ls, n);
  agg_edge1_k<<<nb_e, B, 0, stream>>>(t3, dinv, src, dst, ls, nE);
}